// FDoG_53738630807793
// MI455X (gfx1250) — compile-verified
//
#include <hip/hip_runtime.h>
#include <cstdint>

typedef uint32_t u32;
typedef uint64_t u64;
typedef u32 u32x4 __attribute__((ext_vector_type(4)));
typedef int i32x4 __attribute__((ext_vector_type(4)));
typedef int i32x8 __attribute__((ext_vector_type(8)));

#define NB     4
#define XD     512
#define YD     512
#define PAD    12
#define PSTR   536                     // padded row stride (elements)
#define PPLANE (PSTR * PSTR)           // 287296 elements per padded plane
#define NPIX   (NB * XD * YD)          // 1048576
#define TILE   16
#define HALO   5                        // ETF tap radius (MU)
#define TW     (TILE + 2 * HALO)        // 26  (ETF window)
#define HTR    9                        // trace gather radius (MAX_S steps of <=1px)
#define TWT    (TILE + 2 * HTR)         // 34  (trace window)
#define HDG    3                        // DoG tap radius (MAX_T)
#define TWD    (TILE + 2 * HDG)         // 22  (DoG window)

// ---- workspace layout (in floats) ----
static constexpr size_t OFF_MAX = 0;                              // 1 uint (max |grad|), padded to 16
static constexpr size_t OFF_MAG = 16;                             // NB padded planes
static constexpr size_t OFF_TA0 = OFF_MAG + (size_t)NB * PPLANE;  // tang ping-pong A (2 fields)
static constexpr size_t OFF_TA1 = OFF_TA0 + (size_t)NB * PPLANE;
static constexpr size_t OFF_TB0 = OFF_TA1 + (size_t)NB * PPLANE;  // tang ping-pong B
static constexpr size_t OFF_TB1 = OFF_TB0 + (size_t)NB * PPLANE;
static constexpr size_t OFF_DOG = OFF_TB1 + (size_t)NB * PPLANE;  // NPIX floats, unpadded
static constexpr size_t ZERO_BYTES = OFF_DOG * sizeof(float);     // pads + max must be zeroed

__device__ __forceinline__ float gpdf(float v, float sig) {
  return expf(-v * v / (2.0f * sig * sig)) * 0.3989422804014327f / sig;
}

// ---------- CDNA5 Tensor Data Mover: DMA one 2D fp32 tile (tile_w x tile_h) into LDS ----------
__device__ __forceinline__ void tdm_load_tile_2d(u32 lds_byte, u64 gaddr,
                                                 int tile_w, int tile_h, int stride_elems) {
  // D# group 0: count=1 user descriptor | lds byte addr | 57-bit global addr | type=2 ("image")
  u32x4 g0;
  g0[0] = 1u;
  g0[1] = lds_byte;
  g0[2] = (u32)(gaddr & 0xffffffffull);
  g0[3] = (u32)((gaddr >> 32) & 0x1ffffffull) | (2u << 30);
  // D# group 1: data_size=4B; big tensor dims (source windows are always fully in-bounds);
  // tile_dim0 = contiguous width, tile_dim1 = rows, tensor_dim0_stride = row pitch in elements.
  const u32 td0 = 1u << 20, td1 = 1u << 20;
  i32x8 g1;
  g1[0] = (int)(2u << 16);
  g1[1] = (int)((td0 & 0xffffu) << 16);
  g1[2] = (int)(((td0 >> 16) & 0xffffu) | ((td1 & 0xffffu) << 16));
  g1[3] = (int)(((td1 >> 16) & 0xffffu) | (((u32)tile_w & 0xffffu) << 16));
  g1[4] = (int)((u32)tile_h & 0xffffu);
  g1[5] = (int)(u32)stride_elems;
  g1[6] = 0;
  g1[7] = 0;
  i32x4 z4 = {0, 0, 0, 0};
#if __clang_major__ >= 23
  i32x8 z8 = {0, 0, 0, 0, 0, 0, 0, 0};
  __builtin_amdgcn_tensor_load_to_lds(g0, g1, z4, z4, z8, 0);
#else
  __builtin_amdgcn_tensor_load_to_lds(g0, g1, z4, z4, 0);
#endif
}

__device__ __forceinline__ int iclamp(int v, int lo, int hi) {
  return v < lo ? lo : (v > hi ? hi : v);
}

// ---------- 1: Sobel -> mag (raw) + unit tangent, and global max(mag) ----------
__global__ void fdog_sobel_kernel(const float* __restrict__ img,
                                  float* __restrict__ mag,
                                  float* __restrict__ t0, float* __restrict__ t1,
                                  unsigned* __restrict__ maxp) {
  int idx = blockIdx.x * 256 + threadIdx.x;
  int b = idx >> 18, rem = idx & 0x3ffff, x = rem >> 9, y = rem & 511;
  const float* p = img + (size_t)b * (XD * YD);
  auto at = [&](int xx, int yy) -> float {
    return (xx < 0 || xx > XD - 1 || yy < 0 || yy > YD - 1) ? 0.0f : p[xx * YD + yy];
  };
  float r0 = at(x - 1, y - 1), r1 = at(x - 1, y), r2 = at(x - 1, y + 1);
  float r3 = at(x, y - 1), r5 = at(x, y + 1);
  float r6 = at(x + 1, y - 1), r7 = at(x + 1, y), r8 = at(x + 1, y + 1);
  float gx = (r6 + 2.0f * r7 + r8) - (r0 + 2.0f * r1 + r2);  // d/dx (rows)
  float gy = (r2 + 2.0f * r5 + r8) - (r0 + 2.0f * r3 + r6);  // d/dy (cols)
  float m = sqrtf(gx * gx + gy * gy);
  float inv = (m == 0.0f) ? 0.0f : 1.0f / m;                 // safe_norm
  size_t o = (size_t)b * PPLANE + (size_t)(x + PAD) * PSTR + (y + PAD);
  mag[o] = m;
  t0[o] = -gy * inv;  // tang = normalize(-sob_y, sob_x)
  t1[o] = gx * inv;
  __shared__ float red[256];
  red[threadIdx.x] = m;
  __syncthreads();
  for (int s = 128; s > 0; s >>= 1) {
    if ((int)threadIdx.x < s) red[threadIdx.x] = fmaxf(red[threadIdx.x], red[threadIdx.x + s]);
    __syncthreads();
  }
  if (threadIdx.x == 0) atomicMax(maxp, __float_as_uint(red[0]));  // mag >= 0: uint order == float order
}

// ---------- 2: one ETF smoothing pass (DIRH=0: taps along X rows; DIRH=1: along Y cols) ----------
template <int DIRH>
__global__ void fdog_etf_kernel(const float* __restrict__ mag,
                                const float* __restrict__ st0, const float* __restrict__ st1,
                                float* __restrict__ dt0, float* __restrict__ dt1,
                                const float* __restrict__ maxp) {
  __shared__ float s_m[TW * TW];
  __shared__ float s_a[TW * TW];
  __shared__ float s_b[TW * TW];
  int ty = blockIdx.x, tx = blockIdx.y, b = blockIdx.z;
  int row0 = tx * TILE + PAD - HALO;  // >= PAD-HALO >= 0, tile fully inside padded plane
  int col0 = ty * TILE + PAD - HALO;
  size_t ofs = (size_t)b * PPLANE + (size_t)row0 * PSTR + col0;
  int tid = threadIdx.y * TILE + threadIdx.x;
  if (tid < 32) {  // wave 0 issues the three tile DMAs (TDM ignores EXEC; once per wave)
    tdm_load_tile_2d((u32)(uintptr_t)(void*)s_m, (u64)(uintptr_t)(mag + ofs), TW, TW, PSTR);
    tdm_load_tile_2d((u32)(uintptr_t)(void*)s_a, (u64)(uintptr_t)(st0 + ofs), TW, TW, PSTR);
    tdm_load_tile_2d((u32)(uintptr_t)(void*)s_b, (u64)(uintptr_t)(st1 + ofs), TW, TW, PSTR);
    __builtin_amdgcn_s_wait_tensorcnt(0);
  }
  __syncthreads();

  float invm = 1.0f / *maxp;  // global-max mag normalization folded into the tanh argument
  int lx = threadIdx.y, ly = threadIdx.x;
  int cc = (lx + HALO) * TW + (ly + HALO);
  float mc = s_m[cc], t0c = s_a[cc], t1c = s_b[cc];
  float acc0 = 0.0f, acc1 = 0.0f;
#pragma unroll
  for (int d = -HALO; d <= HALO; ++d) {
    int nn = DIRH ? (cc + d) : (cc + d * TW);
    float mn = s_m[nn], t0n = s_a[nn], t1n = s_b[nn];
    float w = (tanhf((mn - mc) * invm) + 1.0f) * (t0c * t0n + t1c * t1n) * 0.5f;
    acc0 += t0n * w;
    acc1 += t1n * w;
  }
  float nrm = sqrtf(acc0 * acc0 + acc1 * acc1);
  float inv = (nrm == 0.0f) ? 1.0f : 1.0f / nrm;  // safe_norm
  size_t o = (size_t)b * PPLANE + (size_t)(tx * TILE + lx + PAD) * PSTR + (ty * TILE + ly + PAD);
  dt0[o] = acc0 * inv;
  dt1[o] = acc1 * inv;
}

// ---------- 3: DoG along the perpendicular of the flow (7 taps; image tile staged via TDM) ----------
__global__ void fdog_dog_kernel(const float* __restrict__ img,
                                const float* __restrict__ t0, const float* __restrict__ t1,
                                float* __restrict__ dog) {
  __shared__ float s_img[TWD * TWD];
  int ty = blockIdx.x, tx = blockIdx.y, b = blockIdx.z;
  // clamped-origin window: all (rounded, clamped) taps provably land inside [o, o+TWD)
  int oR = iclamp(tx * TILE - HDG, 0, XD - TWD);
  int oC = iclamp(ty * TILE - HDG, 0, YD - TWD);
  const float* p = img + (size_t)b * (XD * YD);
  int tid = threadIdx.y * TILE + threadIdx.x;
  if (tid < 32) {
    tdm_load_tile_2d((u32)(uintptr_t)(void*)s_img,
                     (u64)(uintptr_t)(p + (size_t)oR * YD + oC), TWD, TWD, YD);
    __builtin_amdgcn_s_wait_tensorcnt(0);
  }
  __syncthreads();

  int x = tx * TILE + threadIdx.y, y = ty * TILE + threadIdx.x;
  size_t o = (size_t)b * PPLANE + (size_t)(x + PAD) * PSTR + (y + PAD);
  float per0 = -t1[o], per1 = t0[o];  // perpendicular of etf
  float acc = 0.0f, tw = 0.0f;
#pragma unroll
  for (int t = -HDG; t <= HDG; ++t) {
    float fx = (float)x + per0 * (float)t;
    float fy = (float)y + per1 * (float)t;
    int px = (int)rintf(fminf(fmaxf(fx, 0.0f), (float)(XD - 1)));
    int py = (int)rintf(fminf(fmaxf(fy, 0.0f), (float)(YD - 1)));
    float gw = gpdf((float)t, 1.0f) - 0.99f * gpdf((float)t, 1.6f);
    acc += s_img[(px - oR) * TWD + (py - oC)] * gw;
    tw += gw;
  }
  dog[(size_t)b * (XD * YD) + x * YD + y] = acc / tw;
}

// ---------- 4: FDoG streamline integration + threshold (etf + dog tiles staged via TDM) ----------
__global__ void fdog_trace_kernel(const float* __restrict__ dog,
                                  const float* __restrict__ t0, const float* __restrict__ t1,
                                  int* __restrict__ out) {
  __shared__ float s_d[TWT * TWT];
  __shared__ float s_e0[TWT * TWT];
  __shared__ float s_e1[TWT * TWT];
  int ty = blockIdx.x, tx = blockIdx.y, b = blockIdx.z;
  // etf window in padded planes: PAD(12) >= HTR(9) -> always fully inside, no clamping
  int baseR = tx * TILE - HTR, baseC = ty * TILE - HTR;          // logical window origin (may be <0)
  size_t eofs = (size_t)b * PPLANE + (size_t)(baseR + PAD) * PSTR + (baseC + PAD);
  // dog window (unpadded): clamped origin, provably covers all clamped gather coords
  int oR = iclamp(baseR, 0, XD - TWT);
  int oC = iclamp(baseC, 0, YD - TWT);
  const float* dg = dog + (size_t)b * (XD * YD);
  int tid = threadIdx.y * TILE + threadIdx.x;
  if (tid < 32) {
    tdm_load_tile_2d((u32)(uintptr_t)(void*)s_d,
                     (u64)(uintptr_t)(dg + (size_t)oR * YD + oC), TWT, TWT, YD);
    tdm_load_tile_2d((u32)(uintptr_t)(void*)s_e0, (u64)(uintptr_t)(t0 + eofs), TWT, TWT, PSTR);
    tdm_load_tile_2d((u32)(uintptr_t)(void*)s_e1, (u64)(uintptr_t)(t1 + eofs), TWT, TWT, PSTR);
    __builtin_amdgcn_s_wait_tensorcnt(0);
  }
  __syncthreads();

  int x = tx * TILE + threadIdx.y, y = ty * TILE + threadIdx.x;
  float acc = 0.0f, tw = 0.0f;
  {  // s = 0 tap (dir = +1 branch of the reference)
    float g = gpdf(0.0f, 3.0f);
    acc += s_d[(x - oR) * TWT + (y - oC)] * g;
    tw += g;
  }
#pragma unroll
  for (int dir = -1; dir <= 1; dir += 2) {
    int px = x, py = y;
#pragma unroll
    for (int s = 1; s <= 9; ++s) {
      int ei = (px - baseR) * TWT + (py - baseC);   // |p - tile0| <= s-1+? <= HTR-1 before last step
      float e0 = s_e0[ei], e1 = s_e1[ei];
      float fx = (float)px + e0 * (float)dir;
      float fy = (float)py + e1 * (float)dir;
      px = (int)rintf(fminf(fmaxf(fx, 0.0f), (float)(XD - 1)));
      py = (int)rintf(fminf(fmaxf(fy, 0.0f), (float)(YD - 1)));
      float g = gpdf((float)s, 3.0f);
      acc += s_d[(px - oR) * TWT + (py - oC)] * g;
      tw += g;
    }
  }
  float f = acc / tw;
  out[(size_t)b * (XD * YD) + x * YD + y] = ((f < 0.0f) && (1.0f + tanhf(f) < 0.7f)) ? 0 : 1;
}

extern "C" void kernel_launch(void* const* d_in, const int* in_sizes, int n_in,
                              void* d_out, int out_size, void* d_ws, size_t ws_size,
                              hipStream_t stream) {
  const float* img = (const float*)d_in[0];
  float* w = (float*)d_ws;
  (void)in_sizes; (void)n_in; (void)out_size; (void)ws_size;

  // zero the pads, ping-pong buffers and the running max (deterministic per call)
  hipMemsetAsync(d_ws, 0, ZERO_BYTES, stream);

  dim3 b1(256), g1((NPIX + 255) / 256);
  fdog_sobel_kernel<<<g1, b1, 0, stream>>>(img, w + OFF_MAG, w + OFF_TA0, w + OFF_TA1,
                                           (unsigned*)(w + OFF_MAX));

  dim3 gT(YD / TILE, XD / TILE, NB), bT(TILE, TILE);
  const float* mx = w + OFF_MAX;
  for (int it = 0; it < 3; ++it) {
    fdog_etf_kernel<0><<<gT, bT, 0, stream>>>(w + OFF_MAG, w + OFF_TA0, w + OFF_TA1,
                                              w + OFF_TB0, w + OFF_TB1, mx);
    fdog_etf_kernel<1><<<gT, bT, 0, stream>>>(w + OFF_MAG, w + OFF_TB0, w + OFF_TB1,
                                              w + OFF_TA0, w + OFF_TA1, mx);
  }

  fdog_dog_kernel<<<gT, bT, 0, stream>>>(img, w + OFF_TA0, w + OFF_TA1, w + OFF_DOG);
  fdog_trace_kernel<<<gT, bT, 0, stream>>>(w + OFF_DOG, w + OFF_TA0, w + OFF_TA1, (int*)d_out);
}